// SparseArch_82497731821573
// MI455X (gfx1250) — compile-verified
//
#include <hip/hip_runtime.h>

// SparseArch pooled embedding lookup for MI455X (gfx1250).
// Data-movement bound (~46 MB @ 23.3 TB/s ~= 2us floor). Uses the CDNA5
// Tensor Data Mover in gather mode (8 embedding rows per descriptor DMA'd
// to LDS, TENSORcnt-tracked) and V_WMMA_F32_16X16X4_F32 with an all-ones
// A matrix to do the 4-rows-at-a-time sum pooling.

typedef unsigned int u32;
typedef u32   u32x4 __attribute__((ext_vector_type(4)));
typedef int   i32x4 __attribute__((ext_vector_type(4)));
typedef int   i32x8 __attribute__((ext_vector_type(8)));
typedef float f32x2 __attribute__((ext_vector_type(2)));
typedef float f32x8 __attribute__((ext_vector_type(8)));

#define EMB_DIM 64

__global__ __launch_bounds__(32) void embbag_pool_tdm_wmma(
    const float* __restrict__ table1, const float* __restrict__ table2,
    const int* __restrict__ values1, const int* __restrict__ values2,
    const int* __restrict__ seg1,    const int* __restrict__ seg2,
    float* __restrict__ out, int L, int num_rows)
{
  const int bag  = blockIdx.x;
  const int feat = blockIdx.y;
  const float* __restrict__ table  = feat ? table2  : table1;
  const int*   __restrict__ values = feat ? values2 : values1;
  const int*   __restrict__ seg    = feat ? seg2    : seg1;
  const int lane = threadIdx.x;          // wave32

  // seg is sorted: find [start,end) = rows belonging to this bag.
  int lo = 0, hi = L;
  while (lo < hi) { int mid = (lo + hi) >> 1; if (seg[mid] <  bag) lo = mid + 1; else hi = mid; }
  const int start = lo;
  hi = L;
  while (lo < hi) { int mid = (lo + hi) >> 1; if (seg[mid] <= bag) lo = mid + 1; else hi = mid; }
  const int end = lo;

  __shared__ __align__(128) float lds[8 * EMB_DIM];   // 8 gathered rows, 2KB
  const u32 lds_base = (u32)(uintptr_t)(&lds[0]);     // low 32 bits = LDS offset
  const u32 tlo = (u32)((uintptr_t)table);
  const u32 thi = (u32)(((uintptr_t)table) >> 32);

  f32x8 acc0 = {}, acc1 = {}, acc2 = {}, acc3 = {};
  const f32x2 ones = {1.0f, 1.0f};      // A matrix = all ones (row summation)
  const int half = lane >> 4;           // 0: lanes 0-15, 1: lanes 16-31
  const int col  = lane & 15;           // N within a 16-wide tile

  for (int base = start; base < end; base += 8) {
    int n = end - base; if (n > 8) n = 8;   // rows this chunk (uniform)

    // Unconditional (clamped) index fetch: lanes >= n load a duplicate valid
    // index; the descriptor's tile_dim1 = n makes the extras inert.
    int a = base + (lane & 7);
    if (a >= end) a = end - 1;
    const int myv = values[a];

    i32x4 g2, g3;
    g2[0] = __builtin_amdgcn_readlane(myv, 0);
    g2[1] = __builtin_amdgcn_readlane(myv, 1);
    g2[2] = __builtin_amdgcn_readlane(myv, 2);
    g2[3] = __builtin_amdgcn_readlane(myv, 3);
    g3[0] = __builtin_amdgcn_readlane(myv, 4);
    g3[1] = __builtin_amdgcn_readlane(myv, 5);
    g3[2] = __builtin_amdgcn_readlane(myv, 6);
    g3[3] = __builtin_amdgcn_readlane(myv, 7);

    // ---- Tensor DMA descriptor, gather mode, 32-bit indices ----
    // Group 0: count=1 | gather_index_size(32b)=1<<30 | gather_mode=1<<31,
    //          lds_addr, global_addr[56:0], type=2 in [127:126].
    u32x4 g0;
    g0[0] = 0xC0000001u;
    g0[1] = lds_base;
    g0[2] = tlo;
    g0[3] = (thi & 0x01FFFFFFu) | (2u << 30);

    // Group 1: data_size=4B (2<<16); tensor_dim0 = 64 (bits 79:48);
    //          tensor_dim1 = num_rows (bits 111:80); tile_dim0 = 64 (127:112);
    //          tile_dim1 = n valid indices (143:128); dim0_stride = 64 (207:160).
    i32x8 g1;
    g1[0] = (int)(2u << 16);
    g1[1] = (int)((u32)EMB_DIM << 16);
    g1[2] = (int)(((u32)num_rows & 0xFFFFu) << 16);
    g1[3] = (int)(((u32)EMB_DIM << 16) | ((u32)num_rows >> 16));
    g1[4] = n;
    g1[5] = EMB_DIM;
    g1[6] = 0;
    g1[7] = 0;

    // Extra 6th-arg group required by this toolchain's builtin arity
    // (uncharacterized; zero-filled as in the probe).
    i32x8 gx = {0, 0, 0, 0, 0, 0, 0, 0};

    // WAR guard: previous iteration's LDS reads must finish before this
    // iteration's LDS writes (tail zeroing + TDM) overwrite the buffer.
    asm volatile("s_wait_dscnt 0" ::: "memory");

    // Partial chunk (only the last one per bag): zero tail rows so the
    // WMMA reads are unconditional -- no per-load EXEC predication.
    if (n < 8) {                         // uniform branch
      for (int r = n; r < 8; ++r) {
        lds[r * EMB_DIM + lane]      = 0.0f;
        lds[r * EMB_DIM + 32 + lane] = 0.0f;
      }
    }

    __builtin_amdgcn_tensor_load_to_lds(g0, g1, g2, g3, gx, 0);
    __builtin_amdgcn_s_wait_tensorcnt(0);
    asm volatile("" ::: "memory");   // TDM wrote LDS behind the compiler's back

    // ---- Sum the 8 staged rows with WMMA: D = ones(16x4) * B(4x16) + C ----
    // All 16 output rows are identical column sums; tail rows are zeroed.
    // 2 K-groups of 4 rows x 4 N-tiles of 16 dims.
#pragma unroll
    for (int g = 0; g < 2; ++g) {
      const float* rowa = &lds[(g * 4 + half) * EMB_DIM];       // rows g*4+{0,1}
      const float* rowb = &lds[(g * 4 + 2 + half) * EMB_DIM];   // rows g*4+{2,3}
      f32x2 b0 = { rowa[col +  0], rowb[col +  0] };
      f32x2 b1 = { rowa[col + 16], rowb[col + 16] };
      f32x2 b2 = { rowa[col + 32], rowb[col + 32] };
      f32x2 b3 = { rowa[col + 48], rowb[col + 48] };
      acc0 = __builtin_amdgcn_wmma_f32_16x16x4_f32(false, ones, false, b0, (short)0, acc0, false, false);
      acc1 = __builtin_amdgcn_wmma_f32_16x16x4_f32(false, ones, false, b1, (short)0, acc1, false, false);
      acc2 = __builtin_amdgcn_wmma_f32_16x16x4_f32(false, ones, false, b2, (short)0, acc2, false, false);
      acc3 = __builtin_amdgcn_wmma_f32_16x16x4_f32(false, ones, false, b3, (short)0, acc3, false, false);
    }
  }

  // D VGPR0, lanes 0-15 = row M=0, N=0..15 -> the pooled sums per column.
  if (lane < 16) {
    float* o = out + (size_t)bag * (2 * EMB_DIM) + (size_t)feat * EMB_DIM;
    o[ 0 + lane] = acc0[0];
    o[16 + lane] = acc1[0];
    o[32 + lane] = acc2[0];
    o[48 + lane] = acc3[0];
  }
}

extern "C" void kernel_launch(void* const* d_in, const int* in_sizes, int n_in,
                              void* d_out, int out_size, void* d_ws, size_t ws_size,
                              hipStream_t stream) {
  const float* t1 = (const float*)d_in[0];
  const float* t2 = (const float*)d_in[1];
  const int*   v1 = (const int*)d_in[2];
  const int*   v2 = (const int*)d_in[3];
  const int*   s1 = (const int*)d_in[4];
  const int*   s2 = (const int*)d_in[5];
  float* out = (float*)d_out;

  const int L        = in_sizes[2];                 // jagged values per feature
  const int B        = out_size / (2 * EMB_DIM);    // bags
  const int num_rows = in_sizes[0] / EMB_DIM;       // table rows

  dim3 grid((unsigned)B, 2);
  embbag_pool_tdm_wmma<<<grid, 32, 0, stream>>>(t1, t2, v1, v2, s1, s2, out, L, num_rows);
}